// TriplePatternPooling_15178414424735
// MI455X (gfx1250) — compile-verified
//
#include <hip/hip_runtime.h>

#define NUM_NODES  1048576
#define IN_CH      128
#define OUT_CH     128
#define NUM_GRAPHS 8192

typedef __attribute__((ext_vector_type(2))) float v2f;
typedef __attribute__((ext_vector_type(8))) float v8f;

__device__ inline void atomicMaxF32(float* addr, float val) {
    unsigned int* ua = (unsigned int*)addr;
    unsigned int old = *ua, assumed;
    while (true) {
        float cur = __uint_as_float(old);
        if (cur >= val) break;
        assumed = old;
        old = atomicCAS(ua, assumed, __float_as_uint(val));
        if (old == assumed) break;
    }
}

// ---- init: seg_max = -inf, seg_sum = 0, pooled = 0 --------------------------
__global__ void init_kernel(float* __restrict__ seg_max, float* __restrict__ seg_sum,
                            float* __restrict__ pooled) {
    int i = blockIdx.x * blockDim.x + threadIdx.x;
    if (i < NUM_GRAPHS) { seg_max[i] = -__builtin_inff(); seg_sum[i] = 0.0f; }
    if (i < NUM_GRAPHS * IN_CH) pooled[i] = 0.0f;
}

// ---- pass 1: wave-per-node dot(x[n], attn_w) + atomic segment max -----------
__global__ void score_kernel(const float* __restrict__ x, const long long* __restrict__ batch,
                             const float* __restrict__ attn_w, const float* __restrict__ attn_b,
                             float* __restrict__ scores, float* __restrict__ seg_max) {
    const int lane = threadIdx.x & 31;
    const int node = blockIdx.x * (blockDim.x >> 5) + (threadIdx.x >> 5);
    if (node >= NUM_NODES) return;
    const float4 xv = ((const float4*)(x + (size_t)node * IN_CH))[lane];
    const float4 wv = ((const float4*)attn_w)[lane];
    float p = xv.x * wv.x + xv.y * wv.y + xv.z * wv.z + xv.w * wv.w;
    #pragma unroll
    for (int off = 16; off > 0; off >>= 1) p += __shfl_down(p, off, 32);
    if (lane == 0) {
        float s = p + attn_b[0];
        scores[node] = s;
        atomicMaxF32(&seg_max[(int)batch[node]], s);
    }
}

// ---- exp(s - segmax) + segment sum (wave-coalesced atomics) -----------------
__global__ void exp_kernel(const long long* __restrict__ batch, const float* __restrict__ seg_max,
                           float* __restrict__ scores, float* __restrict__ seg_sum) {
    const int n = blockIdx.x * blockDim.x + threadIdx.x;
    if (n >= NUM_NODES) return;
    const int seg = (int)batch[n];
    float e = __expf(scores[n] - seg_max[seg]);
    scores[n] = e;
    const int seg0 = __shfl(seg, 0, 32);
    if (__all(seg == seg0)) {   // batch is sorted: common case, 1 atomic per wave
        float r = e;
        #pragma unroll
        for (int off = 16; off > 0; off >>= 1) r += __shfl_down(r, off, 32);
        if ((threadIdx.x & 31) == 0) atomicAdd(&seg_sum[seg0], r);
    } else {
        atomicAdd(&seg_sum[seg], e);
    }
}

__global__ void inv_kernel(const float* __restrict__ seg_sum, float* __restrict__ inv_sum) {
    int g = blockIdx.x * blockDim.x + threadIdx.x;
    if (g < NUM_GRAPHS) inv_sum[g] = 1.0f / (seg_sum[g] + 1e-16f);
}

// ---- pass 2: block of 128 threads (1 channel each) over 128 contiguous nodes;
//      register accumulation, atomic flush only at segment boundaries --------
__global__ void pool_kernel(const float* __restrict__ x, const long long* __restrict__ batch,
                            const float* __restrict__ scores, const float* __restrict__ inv_sum,
                            float* __restrict__ pooled) {
    const int c = threadIdx.x;             // 0..127
    const int start = blockIdx.x * 128;
    float acc = 0.0f;
    int prevSeg = (int)batch[start];
    for (int i = 0; i < 128; ++i) {
        const int n = start + i;
        const int seg = (int)batch[n];
        if (seg != prevSeg) {
            atomicAdd(&pooled[(size_t)prevSeg * IN_CH + c], acc);
            acc = 0.0f;
            prevSeg = seg;
        }
        const float w = scores[n] * inv_sum[seg];
        acc += w * x[(size_t)n * IN_CH + c];
    }
    atomicAdd(&pooled[(size_t)prevSeg * IN_CH + c], acc);
}

// ---- final projection: (8192x128) @ (128x128)^T via V_WMMA_F32_16X16X4_F32 --
// One wave per 16x16 output tile. A = pooled tile (16x4 per step),
// B = proj_w rows (proj_w is row-major OUT_CHxIN_CH == column-major B).
__global__ void gemm_kernel(const float* __restrict__ pooled, const float* __restrict__ proj_w,
                            const float* __restrict__ proj_b, float* __restrict__ out) {
    const int wave  = (blockIdx.x * blockDim.x + threadIdx.x) >> 5;
    const int lane  = threadIdx.x & 31;
    const int mtile = wave >> 3;   // 512 tiles of 16 rows
    const int ntile = wave & 7;    // 8 tiles of 16 output channels
    const int r16   = lane & 15;
    const int hi    = lane >> 4;   // lanes 16-31 hold K-pair +2

    const float* arow = pooled + (size_t)(mtile * 16 + r16) * IN_CH;
    const float* brow = proj_w + (size_t)(ntile * 16 + r16) * IN_CH;

    v8f c = {};
    #pragma unroll
    for (int k0 = 0; k0 < IN_CH; k0 += 4) {
        v2f a = *(const v2f*)(arow + k0 + hi * 2);
        v2f b = *(const v2f*)(brow + k0 + hi * 2);
        c = __builtin_amdgcn_wmma_f32_16x16x4_f32(
                /*neg_a=*/false, a, /*neg_b=*/false, b,
                /*c_mod=*/(short)0, c, /*reuse_a=*/false, /*reuse_b=*/false);
    }

    const int col  = ntile * 16 + r16;
    const float bias = proj_b[col];
    #pragma unroll
    for (int r = 0; r < 8; ++r) {
        const int row = mtile * 16 + r + hi * 8;
        out[(size_t)row * OUT_CH + col] = c[r] + bias;
    }
}

extern "C" void kernel_launch(void* const* d_in, const int* in_sizes, int n_in,
                              void* d_out, int out_size, void* d_ws, size_t ws_size,
                              hipStream_t stream) {
    const float*     x      = (const float*)d_in[0];
    // d_in[1] = edge_index (int64), d_in[2] = edge_attr  — unused by reference
    const long long* batch  = (const long long*)d_in[3];
    const float*     attn_w = (const float*)d_in[4];
    const float*     attn_b = (const float*)d_in[5];
    const float*     proj_w = (const float*)d_in[6];
    const float*     proj_b = (const float*)d_in[7];
    float*           out    = (float*)d_out;

    // Workspace layout (bytes)
    char* ws = (char*)d_ws;
    float* scores  = (float*)(ws);                                   // 1M f32 = 4 MB
    float* seg_max = (float*)(ws + (size_t)NUM_NODES * 4);           // 8192 f32
    float* seg_sum = seg_max + NUM_GRAPHS;                           // 8192 f32
    float* inv_sum = seg_sum + NUM_GRAPHS;                           // 8192 f32
    float* pooled  = inv_sum + NUM_GRAPHS;                           // 1M f32 = 4 MB

    // 1) init segment stats + pooled accumulator
    init_kernel<<<(NUM_GRAPHS * IN_CH + 255) / 256, 256, 0, stream>>>(seg_max, seg_sum, pooled);

    // 2) scores + segment max   (wave-per-node: 8 nodes / 256-thread block)
    score_kernel<<<NUM_NODES / 8, 256, 0, stream>>>(x, batch, attn_w, attn_b, scores, seg_max);

    // 3) exp + segment sum
    exp_kernel<<<NUM_NODES / 256, 256, 0, stream>>>(batch, seg_max, scores, seg_sum);

    // 4) reciprocal of segment sums
    inv_kernel<<<(NUM_GRAPHS + 255) / 256, 256, 0, stream>>>(seg_sum, inv_sum);

    // 5) weighted segment pooling
    pool_kernel<<<NUM_NODES / 128, 128, 0, stream>>>(x, batch, scores, inv_sum, pooled);

    // 6) projection GEMM via WMMA: 4096 waves = 512 blocks x 8 waves
    gemm_kernel<<<512, 256, 0, stream>>>(pooled, proj_w, proj_b, out);
}